// BatchNormalizationQ_68788196212990
// MI455X (gfx1250) — compile-verified
//
#include <hip/hip_runtime.h>
#include <hip/hip_bf16.h>

typedef float v2f __attribute__((ext_vector_type(2)));
typedef float v8f __attribute__((ext_vector_type(8)));

#define EPSQ 1e-4f

// ---------------------------------------------------------------------------
// helpers for the WMMA stats kernel
// ---------------------------------------------------------------------------
__device__ __forceinline__ void qbn_load8(const float* __restrict__ x,
                                          uint32_t off, uint32_t C, v2f f[4])
{
    #pragma unroll
    for (int j = 0; j < 4; ++j) {
        uint32_t o = off + (uint32_t)j * 4u * C;   // n advances 4 per fragment
        f[j][0] = x[o];                            // k = kk*2
        f[j][1] = x[o + C];                        // k = kk*2 + 1
    }
}

__device__ __forceinline__ void qbn_wmma4(v8f& acc, float& sum, const v2f f[4])
{
    #pragma unroll
    for (int j = 0; j < 4; ++j) {
        // D(16x16,f32) += A(16x4) * B(4x16); same fragment serves as A and B.
        acc = __builtin_amdgcn_wmma_f32_16x16x4_f32(false, f[j], false, f[j],
                                                    (short)0, acc, false, false);
        sum += f[j][0] + f[j][1];
    }
}

// ---------------------------------------------------------------------------
// Kernel 1: chunked second-moment Gram matrices via V_WMMA_F32_16X16X4_F32.
// grid.x = n-chunks, grid.y = C/32. Block = 256 threads = 8 waves.
// Each wave owns 4 channels; row/col index = c_local*4 + component.
// A fragment == B fragment (lane-for-lane identical layout), so D += X^T X.
// Diagonal 4x4 blocks of D = per-channel component Gram matrices.
// ---------------------------------------------------------------------------
__global__ __launch_bounds__(256)
void qbn_stats_wmma(const float* __restrict__ x,
                    float* __restrict__ partial,   // [NCHUNK][C][24]
                    int C, long long N, int chunkN)
{
    const int tid  = threadIdx.x;
    const int lane = tid & 31;
    const int wave = tid >> 5;
    const int p    = lane & 3;            // component
    const int cl   = (lane & 15) >> 2;    // channel-in-group 0..3
    const int kk   = lane >> 4;           // 0 -> K={0,1}, 1 -> K={2,3}
    const int cg   = blockIdx.y * 32 + wave * 4 + cl;
    // Clamp instead of predicating: OOB-channel lanes only pollute discarded
    // blocks of the 16x16 accumulator, so unconditional loads are safe.
    const int cgc  = (cg < C) ? cg : (C - 1);

    const long long n0 = (long long)blockIdx.x * chunkN;
    long long nEnd = n0 + chunkN; if (nEnd > N) nEnd = N;

    v8f acc = {0.f, 0.f, 0.f, 0.f, 0.f, 0.f, 0.f, 0.f};
    float sum = 0.f;

    const uint32_t uC = (uint32_t)C;
    uint32_t off = ((uint32_t)p * (uint32_t)N + (uint32_t)n0 + (uint32_t)(kk * 2)) * uC
                 + (uint32_t)cgc;

    if (nEnd - n0 == (long long)chunkN) {
        // ---- fast path: full chunk, no bounds checks, software pipelined ----
        const int nOuter = chunkN >> 4;            // 16 n per outer iteration
        v2f f[4];
        qbn_load8(x, off, uC, f);
        off += 16u * uC;
        for (int i = 1; i < nOuter; ++i) {
            v2f g[4];
            qbn_load8(x, off, uC, g);              // prefetch next group
            off += 16u * uC;
            qbn_wmma4(acc, sum, f);                // 4 WMMAs on current group
            #pragma unroll
            for (int j = 0; j < 4; ++j) f[j] = g[j];
        }
        qbn_wmma4(acc, sum, f);
    } else {
        // ---- tail chunk: clamped addresses + selects (no EXEC divergence) ----
        for (long long nb = n0; nb < nEnd; nb += 4) {
            long long n  = nb + (long long)(kk * 2);
            long long nA = (n     < nEnd) ? n       : (nEnd - 1);
            long long nB = (n + 1 < nEnd) ? (n + 1) : (nEnd - 1);
            float v0 = x[((size_t)p * (size_t)N + (size_t)nA) * (size_t)C + (size_t)cgc];
            float v1 = x[((size_t)p * (size_t)N + (size_t)nB) * (size_t)C + (size_t)cgc];
            v0 = (n     < nEnd) ? v0 : 0.f;
            v1 = (n + 1 < nEnd) ? v1 : 0.f;
            v2f frag; frag[0] = v0; frag[1] = v1;
            acc = __builtin_amdgcn_wmma_f32_16x16x4_f32(false, frag, false, frag,
                                                        (short)0, acc, false, false);
            sum += v0 + v1;
        }
    }

    // ---- write partials: [chunk][C][24] = 16 gram + 8 partial sums ----
    size_t chBase = ((size_t)blockIdx.x * (size_t)C + (size_t)cg) * 24;

    // Gram block for column channel cl, column component q=p:
    //   c in {0,1}: rows 0..7  -> lanes 0..15 (kk==0), VGPR = c*4 + prow
    //   c in {2,3}: rows 8..15 -> lanes 16..31 (kk==1), VGPR = (c-2)*4 + prow
    const bool own = ((cl < 2 && kk == 0) || (cl >= 2 && kk == 1));
    if (own && cg < C) {
        const bool hi = (cl & 1);
        float4 g;
        g.x = hi ? acc[4] : acc[0];
        g.y = hi ? acc[5] : acc[1];
        g.z = hi ? acc[6] : acc[2];
        g.w = hi ? acc[7] : acc[3];
        *reinterpret_cast<float4*>(partial + chBase + (size_t)(p * 4)) = g; // [q][prow]
    }
    if (cg < C) {
        partial[chBase + 16 + kk * 4 + p] = sum;   // two half-sums per (c, p)
    }
}

// ---------------------------------------------------------------------------
// Kernel 2: reduce chunk partials, build cov, inverse-Cholesky whitening,
// fuse with gamma, fold mu into beta'.  One thread per channel.
// ---------------------------------------------------------------------------
__global__ __launch_bounds__(256)
void qbn_finalize(const float* __restrict__ partial, // [NCHUNK][C][24]
                  const float* __restrict__ gamma,   // [10][C]
                  const float* __restrict__ beta,    // [4][C]
                  float* __restrict__ Mb,            // [C][20]
                  int C, int nchunk, float invN)
{
    const int c = blockIdx.x * 256 + threadIdx.x;
    if (c >= C) return;

    float g[16]; float s[4];
    #pragma unroll
    for (int i = 0; i < 16; ++i) g[i] = 0.f;
    #pragma unroll
    for (int i = 0; i < 4; ++i) s[i] = 0.f;

    for (int ch = 0; ch < nchunk; ++ch) {
        const float* pc = partial + ((size_t)ch * (size_t)C + (size_t)c) * 24;
        #pragma unroll
        for (int i = 0; i < 16; ++i) g[i] += pc[i];
        #pragma unroll
        for (int pidx = 0; pidx < 4; ++pidx) s[pidx] += pc[16 + pidx] + pc[20 + pidx];
    }

    float mu[4];
    #pragma unroll
    for (int i = 0; i < 4; ++i) mu[i] = s[i] * invN;

    // cov[p][q] = E[x_p x_q] - mu_p mu_q (+EPS diag).  g[q*4+p] = Gram.
    float V[4][4];
    #pragma unroll
    for (int q = 0; q < 4; ++q)
        #pragma unroll
        for (int pidx = 0; pidx < 4; ++pidx) {
            float v = g[q * 4 + pidx] * invN - mu[pidx] * mu[q];
            if (pidx == q) v += EPSQ;
            V[pidx][q] = v;
        }

    // Cholesky factor entries (reference order), then inverse entries.
    float w_rr = sqrtf(V[0][0]);
    float w_ri = V[0][1] / w_rr;
    float w_ii = sqrtf(V[1][1] - w_ri * w_ri);
    float w_rj = V[0][2] / w_rr;
    float w_ij = (V[1][2] - w_ri * w_rj) / w_ii;
    float w_jj = sqrtf(V[2][2] - (w_ij * w_ij + w_rj * w_rj));
    float w_rk = V[0][3] / w_rr;
    float w_ik = (V[1][3] - w_ri * w_rk) / w_ii;
    float w_jk = (V[2][3] - (w_ij * w_ik + w_rj * w_rk)) / w_jj;
    float w_kk = sqrtf(V[3][3] - (w_jk * w_jk + w_ik * w_ik + w_rk * w_rk));

    float o_rr = 1.0f / w_rr;
    float o_ii = 1.0f / w_ii;
    float o_jj = 1.0f / w_jj;
    float o_kk = 1.0f / w_kk;
    float o_ri = -(w_ri * o_rr) / w_ii;
    float o_rj = -(w_rj * o_rr + w_ij * o_ri) / w_jj;
    float o_rk = -(w_rk * o_rr + w_ik * o_ri + w_jk * o_rj) / w_kk;
    float o_ij = -(w_ij * o_ii) / w_jj;
    float o_ik = -(w_ik * o_ii + w_jk * o_ij) / w_kk;
    float o_jk = -(w_jk * o_jj) / w_kk;

    float W[4][4] = {
        { o_rr, o_ri, o_rj, o_rk },
        { o_ri, o_ii, o_ij, o_ik },
        { o_rj, o_ij, o_jj, o_jk },
        { o_rk, o_ik, o_jk, o_kk },
    };

    float g0 = gamma[0 * C + c], g1 = gamma[1 * C + c], g2 = gamma[2 * C + c];
    float g3 = gamma[3 * C + c], g4 = gamma[4 * C + c], g5 = gamma[5 * C + c];
    float g6 = gamma[6 * C + c], g7 = gamma[7 * C + c], g8 = gamma[8 * C + c];
    float g9 = gamma[9 * C + c];
    float G[4][4] = {
        { g0, g1, g2, g3 },
        { g1, g4, g5, g6 },
        { g2, g5, g7, g8 },
        { g3, g6, g8, g9 },
    };

    float* mb = Mb + (size_t)c * 20;
    #pragma unroll
    for (int pidx = 0; pidx < 4; ++pidx) {
        float bp = beta[pidx * C + c];
        #pragma unroll
        for (int sidx = 0; sidx < 4; ++sidx) {
            float m = 0.f;
            #pragma unroll
            for (int t = 0; t < 4; ++t) m += G[pidx][t] * W[t][sidx];
            mb[pidx * 4 + sidx] = m;
            bp -= m * mu[sidx];
        }
        mb[16 + pidx] = bp;
    }
}

// ---------------------------------------------------------------------------
// Kernel 3: out[p,n,c] = sum_s M[p][s][c] * x[s,n,c] + beta'[p][c].
// One channel per thread (coalesced along C), M register-resident,
// u32 offsets, `iters` consecutive n per block.
// ---------------------------------------------------------------------------
__global__ __launch_bounds__(256)
void qbn_apply(const float* __restrict__ x,
               const float* __restrict__ Mb,   // [C][20]
               float* __restrict__ out,
               int C, long long N, int iters)
{
    const int c = blockIdx.y * 256 + threadIdx.x;
    if (c >= C) return;

    float m[16], b[4];
    const float* mb = Mb + (size_t)c * 20;
    #pragma unroll
    for (int i = 0; i < 16; ++i) m[i] = mb[i];
    #pragma unroll
    for (int i = 0; i < 4; ++i) b[i] = mb[16 + i];

    const uint32_t NC = (uint32_t)(N * (long long)C);     // < 2^32 here
    long long nBase = (long long)blockIdx.x * iters;
    long long nStop = nBase + iters; if (nStop > N) nStop = N;

    uint32_t off = (uint32_t)(nBase * (long long)C + (long long)c);
    for (long long n = nBase; n < nStop; ++n, off += (uint32_t)C) {
        float xs0 = x[off];
        float xs1 = x[off + NC];
        float xs2 = x[off + 2u * NC];
        float xs3 = x[off + 3u * NC];
        #pragma unroll
        for (int pidx = 0; pidx < 4; ++pidx) {
            float o = b[pidx];
            o = fmaf(m[pidx * 4 + 0], xs0, o);
            o = fmaf(m[pidx * 4 + 1], xs1, o);
            o = fmaf(m[pidx * 4 + 2], xs2, o);
            o = fmaf(m[pidx * 4 + 3], xs3, o);
            out[off + (uint32_t)pidx * NC] = o;
        }
    }
}

// ---------------------------------------------------------------------------
extern "C" void kernel_launch(void* const* d_in, const int* in_sizes, int n_in,
                              void* d_out, int out_size, void* d_ws, size_t ws_size,
                              hipStream_t stream)
{
    const float* x     = (const float*)d_in[0];   // [4,B,H,W,C]
    const float* gamma = (const float*)d_in[1];   // [10,C]
    const float* beta  = (const float*)d_in[2];   // [4,C]
    float* out = (float*)d_out;

    const int C = in_sizes[1] / 10;
    const long long N = (long long)in_sizes[0] / (4LL * C);   // B*H*W

    // n-chunking for the stats pass (chunk length multiple of 16: 4 WMMAs of K=4)
    int nchunk = 112;
    int chunkN = (int)((N + nchunk - 1) / nchunk);
    chunkN = (chunkN + 15) & ~15;
    nchunk = (int)((N + chunkN - 1) / chunkN);

    // workspace: [C][20] fused matrices, then [nchunk][C][24] partials
    float* Mb      = (float*)d_ws;
    float* partial = Mb + (size_t)C * 20;

    dim3 g1((unsigned)nchunk, (unsigned)((C + 31) / 32));
    qbn_stats_wmma<<<g1, 256, 0, stream>>>(x, partial, C, N, chunkN);

    qbn_finalize<<<(C + 255) / 256, 256, 0, stream>>>(partial, gamma, beta, Mb,
                                                      C, nchunk, 1.0f / (float)N);

    const int iters = 16;
    dim3 g3((unsigned)((N + iters - 1) / iters), (unsigned)((C + 255) / 256));
    qbn_apply<<<g3, 256, 0, stream>>>(x, Mb, out, C, N, iters);
}